// AMNet_10273561772518
// MI455X (gfx1250) — compile-verified
//
#include <hip/hip_runtime.h>
#include <hip/hip_bf16.h>
#include <math.h>

// ---------------------------------------------------------------------------
// AMNet forward for MI455X (gfx1250, wave32).
//   * 3 SpMMs total (algebraic CSE across the 3 BernConv filters)
//   * dense MLP uses V_WMMA_F32_16X16X4_F32 (full-precision f32 matrix path)
//   * segment sums via global_atomic_add_f32 (unsafeAtomicAdd)
// Shapes: N=in_sizes[0]/64 nodes, E edges, F_IN=64, H=32, L=3, K=2, C=2.
// ---------------------------------------------------------------------------

typedef __attribute__((ext_vector_type(2))) float v2f;
typedef __attribute__((ext_vector_type(8))) float v8f;

#define HID 32
#define FIN 64

// ---------------- trivial helpers ----------------
__global__ void zero4_kernel(float4* __restrict__ p, int n4) {
  int i = blockIdx.x * blockDim.x + threadIdx.x;
  if (i < n4) p[i] = make_float4(0.f, 0.f, 0.f, 0.f);
}

__global__ void degree_kernel(const int* __restrict__ dst, float* __restrict__ deg, int E) {
  int e = blockIdx.x * blockDim.x + threadIdx.x;
  if (e < E) unsafeAtomicAdd(&deg[dst[e]], 1.0f);
}

__global__ void dinv_kernel(float* __restrict__ d, int N) {
  int i = blockIdx.x * blockDim.x + threadIdx.x;
  if (i < N) {
    float v = d[i];
    v = v < 1.0f ? 1.0f : v;       // clip(deg, 1, None)
    d[i] = rsqrtf(v);              // deg^-0.5
  }
}

// ---------------- MLP: h = relu(X@W1+b1)@W2+b2, via f32 WMMA ----------------
// 8 waves/block, 16 node-rows per wave -> 128 rows per block.
// Weights are staged in LDS pre-swizzled into WMMA B-fragment order
// [k0/4][col][kk] so each lane reads one aligned b64 per tile per k-step.
// Tail rows are handled by CLAMPING the A row index: in D=A*B a garbage row
// only affects its own output row, and those stores are guarded.
__global__ __launch_bounds__(256) void mlp_wmma_kernel(
    const float* __restrict__ X, const float* __restrict__ W1,
    const float* __restrict__ b1, const float* __restrict__ W2,
    const float* __restrict__ b2, float* __restrict__ h, int N) {
  __shared__ float W1s[FIN / 4][HID][4];   // 8 KB, fragment-swizzled
  __shared__ float W2s[HID / 4][HID][4];   // 4 KB, fragment-swizzled
  __shared__ float b1s[HID], b2s[HID];
  __shared__ float hms[8][16][HID];        // per-wave relu(X@W1+b1) staging, 16 KB

  int tid = threadIdx.x;
  for (int i = tid; i < FIN * HID; i += 256) {
    int k = i >> 5, col = i & 31;
    W1s[k >> 2][col][k & 3] = W1[i];
  }
  for (int i = tid; i < HID * HID; i += 256) {
    int k = i >> 5, col = i & 31;
    W2s[k >> 2][col][k & 3] = W2[i];
  }
  if (tid < HID) { b1s[tid] = b1[tid]; b2s[tid] = b2[tid]; }
  __syncthreads();

  int wave = tid >> 5;
  int lane = tid & 31;
  int half = lane >> 4;                 // lane half selects K-pair / M-offset
  int r    = lane & 15;
  int rowBase = blockIdx.x * 128 + wave * 16;
  int arow = rowBase + r;               // row this lane feeds into A-fragments
  int crow = arow < N ? arow : N - 1;   // clamp (tail rows are never stored)
  const float* Xrow = X + (size_t)crow * FIN + 2 * half;

  // ---- GEMM 1: [16,64] x [64,32] ----
  v8f c0 = {}; v8f c1 = {};
  for (int k0 = 0; k0 < FIN; k0 += 4) {
    v2f a  = *(const v2f*)(Xrow + k0);                 // A: v0=K0/K2, v1=K1/K3
    v2f bA = *(const v2f*)(&W1s[k0 >> 2][r][2 * half]);      // cols 0..15
    v2f bB = *(const v2f*)(&W1s[k0 >> 2][16 + r][2 * half]); // cols 16..31
    c0 = __builtin_amdgcn_wmma_f32_16x16x4_f32(false, a, false, bA, (short)0, c0, false, false);
    c1 = __builtin_amdgcn_wmma_f32_16x16x4_f32(false, a, false, bB, (short)0, c1, false, false);
  }
  // bias + relu, stage to LDS row-major so it can be re-read as A-fragments
#pragma unroll
  for (int i = 0; i < 8; i++) {
    int m = i + 8 * half;               // C/D layout: M = vgpr + 8*(lane>=16)
    hms[wave][m][r]      = fmaxf(c0[i] + b1s[r], 0.0f);
    hms[wave][m][16 + r] = fmaxf(c1[i] + b1s[16 + r], 0.0f);
  }
  __syncthreads();

  // ---- GEMM 2: [16,32] x [32,32] ----
  v8f d0 = {}; v8f d1 = {};
  for (int k0 = 0; k0 < HID; k0 += 4) {
    v2f a  = *(const v2f*)(&hms[wave][r][k0 + 2 * half]);
    v2f bA = *(const v2f*)(&W2s[k0 >> 2][r][2 * half]);
    v2f bB = *(const v2f*)(&W2s[k0 >> 2][16 + r][2 * half]);
    d0 = __builtin_amdgcn_wmma_f32_16x16x4_f32(false, a, false, bA, (short)0, d0, false, false);
    d1 = __builtin_amdgcn_wmma_f32_16x16x4_f32(false, a, false, bB, (short)0, d1, false, false);
  }
#pragma unroll
  for (int i = 0; i < 8; i++) {
    int m = i + 8 * half;
    int row = rowBase + m;
    if (row < N) {
      h[(size_t)row * HID + r]      = d0[i] + b2s[r];
      h[(size_t)row * HID + 16 + r] = d1[i] + b2s[16 + r];
    }
  }
}

// ---------------- SpMM: out[v][ch] += x[u][ch]*dinv[u] over edges ----------------
// One thread per (edge, channel); 32 channels => lane index == channel, so the
// gather of x[u][0..31] and the 32 atomics to out[v][0..31] are both 128B-contiguous.
__global__ void spmm_kernel(const float* __restrict__ x, const float* __restrict__ dinv,
                            const int* __restrict__ src, const int* __restrict__ dst,
                            float* __restrict__ outp, int E) {
  int idx = blockIdx.x * blockDim.x + threadIdx.x;
  int e = idx >> 5, ch = idx & 31;
  if (e < E) {
    int u = src[e], v = dst[e];
    float val = x[(size_t)u * HID + ch] * dinv[u];
    unsafeAtomicAdd(&outp[(size_t)v * HID + ch], val);
  }
}

// t1 = h + dinv*a ; u = h - dinv*a   (a = S(h), shared by lap2 and lap1 chains)
// float4-vectorized: 8 vec4 per node row.
__global__ void stage1_kernel(const float4* __restrict__ h, const float4* __restrict__ acc,
                              const float* __restrict__ dinv, float4* __restrict__ t1,
                              float4* __restrict__ u, int nvec) {
  int i = blockIdx.x * blockDim.x + threadIdx.x;
  if (i < nvec) {
    float dv = dinv[i >> 3];
    float4 a = acc[i], hv = h[i];
    float4 t, uu;
    t.x = fmaf(a.x, dv, hv.x);  uu.x = fmaf(-a.x, dv, hv.x);
    t.y = fmaf(a.y, dv, hv.y);  uu.y = fmaf(-a.y, dv, hv.y);
    t.z = fmaf(a.z, dv, hv.z);  uu.z = fmaf(-a.z, dv, hv.z);
    t.w = fmaf(a.w, dv, hv.w);  uu.w = fmaf(-a.w, dv, hv.w);
    t1[i] = t; u[i] = uu;
  }
}

// b = S(t1): t2 = t1 + dinv*b, y1 = t1 - dinv*b ; F_l = 0.25 w_l0 t2 + 0.5 w_l1 y1
__global__ void stage2_kernel(const float4* __restrict__ t1, const float4* __restrict__ acc,
                              const float* __restrict__ dinv, const float* __restrict__ wbern,
                              float4* __restrict__ F, int nvec) {
  int i = blockIdx.x * blockDim.x + threadIdx.x;
  if (i < nvec) {
    float dv = dinv[i >> 3];
    float4 b = acc[i], tv = t1[i];
    float4 t2, y1;
    t2.x = fmaf(b.x, dv, tv.x);  y1.x = fmaf(-b.x, dv, tv.x);
    t2.y = fmaf(b.y, dv, tv.y);  y1.y = fmaf(-b.y, dv, tv.y);
    t2.z = fmaf(b.z, dv, tv.z);  y1.z = fmaf(-b.z, dv, tv.z);
    t2.w = fmaf(b.w, dv, tv.w);  y1.w = fmaf(-b.w, dv, tv.w);
#pragma unroll
    for (int l = 0; l < 3; l++) {
      float w0 = 0.25f * fmaxf(wbern[l * 3 + 0], 0.0f);   // relu(w) * C(2,0)/4
      float w1 = 0.50f * fmaxf(wbern[l * 3 + 1], 0.0f);   // relu(w) * C(2,1)/4
      float4 o;
      o.x = w0 * t2.x + w1 * y1.x;
      o.y = w0 * t2.y + w1 * y1.y;
      o.z = w0 * t2.z + w1 * y1.z;
      o.w = w0 * t2.w + w1 * y1.w;
      F[(size_t)l * nvec + i] = o;
    }
  }
}

// c = S(u): y2 = u - dinv*c ; F_l += 0.25 w_l2 y2
__global__ void stage3_kernel(const float4* __restrict__ u, const float4* __restrict__ acc,
                              const float* __restrict__ dinv, const float* __restrict__ wbern,
                              float4* __restrict__ F, int nvec) {
  int i = blockIdx.x * blockDim.x + threadIdx.x;
  if (i < nvec) {
    float dv = dinv[i >> 3];
    float4 c = acc[i], uv = u[i];
    float4 y2;
    y2.x = fmaf(-c.x, dv, uv.x);
    y2.y = fmaf(-c.y, dv, uv.y);
    y2.z = fmaf(-c.z, dv, uv.z);
    y2.w = fmaf(-c.w, dv, uv.w);
#pragma unroll
    for (int l = 0; l < 3; l++) {
      float w2 = 0.25f * fmaxf(wbern[l * 3 + 2], 0.0f);   // relu(w) * C(2,2)/4
      float4 o = F[(size_t)l * nvec + i];
      o.x = fmaf(w2, y2.x, o.x);
      o.y = fmaf(w2, y2.y, o.y);
      o.z = fmaf(w2, y2.z, o.z);
      o.w = fmaf(w2, y2.w, o.w);
      F[(size_t)l * nvec + i] = o;
    }
  }
}

// ---------------- attention + classifier: one wave32 per node ----------------
__global__ __launch_bounds__(256) void attention_kernel(
    const float* __restrict__ h, const float* __restrict__ F,
    const float* __restrict__ W2, const float* __restrict__ b2,
    const float* __restrict__ Wc, const float* __restrict__ bc,
    float* __restrict__ out, int N) {
  __shared__ float W2s[HID * HID];
  __shared__ float b2s[HID], Wcs[HID * 2], bcs[2];
  int tid = threadIdx.x;
  for (int i = tid; i < HID * HID; i += 256) W2s[i] = W2[i];
  if (tid < HID) b2s[tid] = b2[tid];
  if (tid < HID * 2) Wcs[tid] = Wc[tid];
  if (tid < 2) bcs[tid] = bc[tid];
  __syncthreads();

  int wave = tid >> 5, j = tid & 31;    // j = hidden channel (H == warpSize == 32)
  int n = blockIdx.x * 8 + wave;
  if (n >= N) return;

  size_t NN = (size_t)N * HID;
  float hj = h[(size_t)n * HID + j];
  float f0 = F[(size_t)n * HID + j];
  float f1 = F[NN + (size_t)n * HID + j];
  float f2 = F[2 * NN + (size_t)n * HID + j];

  // projections through W2: column j accumulated via lane broadcasts
  float xp = b2s[j], p0 = b2s[j], p1 = b2s[j], p2 = b2s[j];
  for (int i = 0; i < HID; i++) {
    float w = W2s[i * HID + j];
    xp = fmaf(__shfl(hj, i), w, xp);
    p0 = fmaf(__shfl(f0, i), w, p0);
    p1 = fmaf(__shfl(f1, i), w, p1);
    p2 = fmaf(__shfl(f2, i), w, p2);
  }
  xp = tanhf(xp); p0 = tanhf(p0); p1 = tanhf(p1); p2 = tanhf(p2);

  // logits = <proj_l, x_proj>  (butterfly reduction over 32 lanes)
  float l0 = p0 * xp, l1 = p1 * xp, l2 = p2 * xp;
  for (int off = 16; off > 0; off >>= 1) {
    l0 += __shfl_xor(l0, off);
    l1 += __shfl_xor(l1, off);
    l2 += __shfl_xor(l2, off);
  }
  // softmax over 3 filters (computed redundantly on all lanes)
  float m = fmaxf(l0, fmaxf(l1, l2));
  float e0 = expf(l0 - m), e1 = expf(l1 - m), e2 = expf(l2 - m);
  float inv = 1.0f / (e0 + e1 + e2);
  float res = (e0 * f0 + e1 * f1 + e2 * f2) * inv;

  // classifier: out[n] = res @ Wc + bc
  float o0 = res * Wcs[j * 2 + 0];
  float o1 = res * Wcs[j * 2 + 1];
  for (int off = 16; off > 0; off >>= 1) {
    o0 += __shfl_xor(o0, off);
    o1 += __shfl_xor(o1, off);
  }
  if (j == 0) {
    out[(size_t)n * 2 + 0] = o0 + bcs[0];
    out[(size_t)n * 2 + 1] = o1 + bcs[1];
  }
}

// ---------------------------------------------------------------------------
static inline int cdiv_ll(long long a, long long b) { return (int)((a + b - 1) / b); }

extern "C" void kernel_launch(void* const* d_in, const int* in_sizes, int n_in,
                              void* d_out, int out_size, void* d_ws, size_t ws_size,
                              hipStream_t stream) {
  const float* feature = (const float*)d_in[0];
  const float* W1      = (const float*)d_in[1];
  const float* b1      = (const float*)d_in[2];
  const float* W2      = (const float*)d_in[3];
  const float* b2      = (const float*)d_in[4];
  const float* Wc      = (const float*)d_in[5];
  const float* bc      = (const float*)d_in[6];
  const float* wbern   = (const float*)d_in[7];
  const int*   src     = (const int*)d_in[8];
  const int*   dst     = (const int*)d_in[9];

  int N = in_sizes[0] / FIN;
  int E = in_sizes[8];
  int nvec = N * (HID / 4);            // float4 count per [N,32] buffer

  // workspace layout (floats): dinv | h | acc | t1 | u | F[3]  => 225*N floats
  float* ws   = (float*)d_ws;
  float* dinv = ws;
  float* h    = dinv + N;
  float* acc  = h   + (size_t)N * HID;
  float* t1   = acc + (size_t)N * HID;
  float* u    = t1  + (size_t)N * HID;
  float* F    = u   + (size_t)N * HID;

  dim3 blk(256);
  long long sp = (long long)E * HID;

  // degrees -> D^{-1/2}   (zero N floats = N/4 float4, N % 4 == 0 here; pad-safe ceil)
  zero4_kernel<<<cdiv_ll((N + 3) / 4, 256), blk, 0, stream>>>((float4*)dinv, (N + 3) / 4);
  degree_kernel<<<cdiv_ll(E, 256), blk, 0, stream>>>(dst, dinv, E);
  dinv_kernel<<<cdiv_ll(N, 256), blk, 0, stream>>>(dinv, N);

  // h = relu(X@W1+b1)@W2+b2  (f32 WMMA)
  mlp_wmma_kernel<<<cdiv_ll(N, 128), blk, 0, stream>>>(feature, W1, b1, W2, b2, h, N);

  // a = S(h) ; t1 = lap2(h), u = lap1(h)
  zero4_kernel<<<cdiv_ll(nvec, 256), blk, 0, stream>>>((float4*)acc, nvec);
  spmm_kernel<<<cdiv_ll(sp, 256), blk, 0, stream>>>(h, dinv, src, dst, acc, E);
  stage1_kernel<<<cdiv_ll(nvec, 256), blk, 0, stream>>>((const float4*)h, (const float4*)acc,
                                                        dinv, (float4*)t1, (float4*)u, nvec);

  // b = S(t1) ; F_l = 0.25 w0 t2 + 0.5 w1 y1
  zero4_kernel<<<cdiv_ll(nvec, 256), blk, 0, stream>>>((float4*)acc, nvec);
  spmm_kernel<<<cdiv_ll(sp, 256), blk, 0, stream>>>(t1, dinv, src, dst, acc, E);
  stage2_kernel<<<cdiv_ll(nvec, 256), blk, 0, stream>>>((const float4*)t1, (const float4*)acc,
                                                        dinv, wbern, (float4*)F, nvec);

  // c = S(u) ; F_l += 0.25 w2 y2
  zero4_kernel<<<cdiv_ll(nvec, 256), blk, 0, stream>>>((float4*)acc, nvec);
  spmm_kernel<<<cdiv_ll(sp, 256), blk, 0, stream>>>(u, dinv, src, dst, acc, E);
  stage3_kernel<<<cdiv_ll(nvec, 256), blk, 0, stream>>>((const float4*)u, (const float4*)acc,
                                                        dinv, wbern, (float4*)F, nvec);

  // attention over filters + classifier
  attention_kernel<<<cdiv_ll(N, 8), blk, 0, stream>>>(h, F, W2, b2, Wc, bc,
                                                      (float*)d_out, N);
}